// MoEGate_77567109366039
// MI455X (gfx1250) — compile-verified
//
#include <hip/hip_runtime.h>

typedef __attribute__((ext_vector_type(2))) float v2f;
typedef __attribute__((ext_vector_type(4))) float v4f;
typedef __attribute__((ext_vector_type(8))) float v8f;

#define HIDDEN  2048
#define NEXP    5
#define TPB     256                        // 8 waves of 32
#define WAVES   (TPB / 32)
#define TOK_PW  16                         // tokens per wave (one WMMA M-tile)
#define KCHUNK  64                         // K per pipeline stage
#define PITCH   (KCHUNK + 4)               // 68: stride 4 (mod 64) -> conflict-free A reads
#define NSLOT   (TOK_PW * KCHUNK / 4 / 32) // 8 float4 slots per lane per chunk

__global__ __launch_bounds__(TPB) void moe_gate_kernel(
    const float* __restrict__ X,    // [N, HIDDEN] tokens
    const float* __restrict__ W,    // [NEXP, HIDDEN] router weights
    float* __restrict__ outW,       // [N, 2] top-k weights
    float* __restrict__ outI,       // [N, 2] top-k indices (as float)
    float* __restrict__ outT)       // [N, HIDDEN] token copy
{
    // per-wave private LDS regions -> entire kernel is barrier-free
    __shared__ float tile[WAVES][TOK_PW * PITCH];
    __shared__ float lgt[WAVES][16][17];

    const int tid  = threadIdx.x;
    const int wave = tid >> 5;
    const int lane = tid & 31;
    const int hl   = lane >> 4;            // half-wave: K-pair select of 16x16x4
    const int m    = lane & 15;            // A row (token) / B col (expert)
    const long tokW = (long)blockIdx.x * (WAVES * TOK_PW) + (long)wave * TOK_PW;

    // staging geometry: slot s = lane + s*32, 16 float4 per 64-float row
    int rowS[NSLOT], colS[NSLOT];
    #pragma unroll
    for (int s = 0; s < NSLOT; ++s) {
        int idx = lane + s * 32;
        rowS[s] = idx >> 4;
        colS[s] = (idx & 15) << 2;
    }

    // branch-free B-fragment source: clamped expert row + 0/1 mask (no EXEC churn)
    const float  msk = (m < NEXP) ? 1.0f : 0.0f;
    const float* Wl  = W + (m < NEXP ? m : 0) * HIDDEN + hl * 2;
    float* tw = tile[wave];

    v8f acc = {0.f, 0.f, 0.f, 0.f, 0.f, 0.f, 0.f, 0.f};

    // prologue: chunk 0 -> registers (8 outstanding NT b128 loads per lane)
    v4f r[NSLOT];
    #pragma unroll
    for (int s = 0; s < NSLOT; ++s)
        r[s] = __builtin_nontemporal_load(
            (const v4f*)(X + (tokW + rowS[s]) * HIDDEN + colS[s]));

    for (int k0 = 0; k0 < HIDDEN; k0 += KCHUNK) {
        // drain staged registers: wave-private LDS tile + streaming (NT) token copy
        #pragma unroll
        for (int s = 0; s < NSLOT; ++s) {
            v4f v = r[s];
            *(v4f*)(tw + rowS[s] * PITCH + colS[s]) = v;
            __builtin_nontemporal_store(v,
                (v4f*)(outT + (tokW + rowS[s]) * HIDDEN + k0 + colS[s]));
        }

        // issue next chunk's loads -> overlap the whole WMMA phase (no barrier)
        const int k1 = k0 + KCHUNK;
        if (k1 < HIDDEN) {
            #pragma unroll
            for (int s = 0; s < NSLOT; ++s)
                r[s] = __builtin_nontemporal_load(
                    (const v4f*)(X + (tokW + rowS[s]) * HIDDEN + k1 + colS[s]));
        }

        // B fragments: unconditional L2-resident loads, masked past expert 4
        v2f bf[KCHUNK / 4];
        #pragma unroll
        for (int j = 0; j < KCHUNK / 4; ++j) {
            bf[j].x = Wl[k0 + 4 * j]     * msk;
            bf[j].y = Wl[k0 + 4 * j + 1] * msk;
        }

        // logits: 16x V_WMMA_F32_16X16X4_F32; A-frag ds_loads precede next
        // iteration's ds_stores, and same-wave DS ops execute in order -> safe
        const float* arow = tw + m * PITCH + hl * 2;
        #pragma unroll
        for (int j = 0; j < KCHUNK / 4; ++j) {
            v2f a;
            a.x = arow[4 * j];
            a.y = arow[4 * j + 1];
            acc = __builtin_amdgcn_wmma_f32_16x16x4_f32(
                false, a, false, bf[j], (short)0, acc, false, false);
        }
    }

    // wave-private logit spill (C layout: VGPR q -> M = q + 8*half, N = lane%16)
    #pragma unroll
    for (int q = 0; q < 8; ++q)
        lgt[wave][q + 8 * hl][m] = acc[q];

    // wave-private epilogue: lanes 0-15 own one token each (in-order DS -> no barrier)
    if (lane < 16) {
        float l[NEXP];
        #pragma unroll
        for (int e = 0; e < NEXP; ++e) l[e] = lgt[wave][lane][e];

        int bi = 0; float bv = l[0];
        #pragma unroll
        for (int e = 1; e < NEXP; ++e)
            if (l[e] > bv) { bv = l[e]; bi = e; }   // strict > : lowest index wins ties
        int si = 0; float sv = -3.0e38f;
        #pragma unroll
        for (int e = 0; e < NEXP; ++e)
            if (e != bi && l[e] > sv) { sv = l[e]; si = e; }

        float rr  = __expf(sv - bv);        // e^{l2-l1}, sv <= bv: no overflow
        float inv = 1.f / (1.f + rr);
        long tok = tokW + lane;
        outW[tok * 2 + 0] = inv;            // p1/(p1+p2)
        outW[tok * 2 + 1] = rr * inv;       // p2/(p1+p2)
        outI[tok * 2 + 0] = (float)bi;
        outI[tok * 2 + 1] = (float)si;
    }
}

extern "C" void kernel_launch(void* const* d_in, const int* in_sizes, int n_in,
                              void* d_out, int out_size, void* d_ws, size_t ws_size,
                              hipStream_t stream) {
    const float* X = (const float*)d_in[0];      // hidden_states [8,4096,2048] f32
    const float* W = (const float*)d_in[1];      // router weight  [5,2048]     f32
    const long N = (long)in_sizes[0] / HIDDEN;   // 32768 tokens

    float* outW = (float*)d_out;      // [N,2] weights
    float* outI = outW + 2 * N;       // [N,2] indices (float-encoded)
    float* outT = outW + 4 * N;       // [N,HIDDEN] token copy

    const long tokPerBlock = WAVES * TOK_PW;     // 128
    const int blocks = (int)((N + tokPerBlock - 1) / tokPerBlock);   // 256
    moe_gate_kernel<<<blocks, TPB, 0, stream>>>(X, W, outW, outI, outT);
}